// TR_Conv_37967510896825
// MI455X (gfx1250) — compile-verified
//
#include <hip/hip_runtime.h>

// ---------------------------------------------------------------------------
// MI455X (gfx1250) fused stride-3 transposed-conv (polyphase) + epilogue.
//   res_rho[co,m] = sum_ci sum_t w[co,ci,3t+rho_k] * x[ci,m+t]   (K=64 per tap)
// V_WMMA_F32_16X16X4_F32 (fp32-exact) with:
//   - K-permuted LDS layout so A/B fragments are contiguous b128 loads
//   - double-buffered (per-tap) register pipeline to hide LDS latency
//   - dual position-tile accumulators to reuse A fragments
// ---------------------------------------------------------------------------

typedef float v2f __attribute__((ext_vector_type(2)));
typedef float v4f __attribute__((ext_vector_type(4)));
typedef float v8f __attribute__((ext_vector_type(8)));

#define T_LEN 8192
#define NB    16
#define NP    128          // output positions per workgroup
#define ME    144          // extended residue positions: 9 tiles of 16, [p0-8, p0+136)
#define XROWS 148          // ME + 4 tap halo
#define STR   68           // padded LDS row stride in floats (bank-conflict avoidance)

// LDS layout (float offsets), dynamic shared: total 61,264 floats = 245,056 B
#define XS_OFF   0
#define W_OFF    (XS_OFF + XROWS * STR)        // x tile: [XROWS][STR]
#define R_OFF0   (W_OFF + 5 * 64 * STR)        // weights: [tap<=5][co 64][STR]
#define RES_SZ   (ME * STR)                    // one residue buffer [ME][STR]
#define BIAS_OFF (R_OFF0 + 3 * RES_SZ)
#define LDS_FLOATS (BIAS_OFF + 64)

// K-slot permutation: lane-half lh reads slots [32*lh, 32*lh+32) contiguously.
// slot s holds ci = 4*((s&31)>>1) + 2*(s>>5) + (s&1); inverse below.
__device__ __forceinline__ int kslot(int ci) {
    return (((ci >> 1) & 1) << 5) + ((ci >> 2) << 1) + (ci & 1);
}

// reflect-pad index map (shared by res0/res1/res2 slices, all length 8188 -> 8192)
__device__ __forceinline__ int ref_idx(int i) {
    if (i < 2) return 2 - i;                           // 0->2, 1->1
    if (i > T_LEN - 3) return (i == T_LEN - 2) ? T_LEN - 6 : T_LEN - 7; // 8190->8186, 8191->8185
    return i - 2;
}

// One (or two) 16x16 output tiles for one residue: full K = 64ci x TAPS taps.
// aw_base = W_OFF + (ct*16+nm)*STR + lh*32          (A: w[co=M][kslot])
// xb_base = XS_OFF + nm*STR + lh*32                 (B: x[pos=N][kslot])
// res_base = R_OFF0 + rho*RES_SZ + nm*STR + ct*16 + lh*8
template<int TAPS, bool DUAL>
__device__ __forceinline__ void gemm_tiles(float* lds, int aw_base, int xb_base,
                                           int res_base, int pt0)
{
    const int x0 = xb_base + pt0 * 16 * STR;
    const int x1 = x0 + 16 * STR;

    v4f A[2][8], B0[2][8], B1[2][8];
    #pragma unroll
    for (int q = 0; q < 8; ++q) {
        A[0][q]  = *(const v4f*)&lds[aw_base + 4 * q];
        B0[0][q] = *(const v4f*)&lds[x0 + 4 * q];
        if constexpr (DUAL) B1[0][q] = *(const v4f*)&lds[x1 + 4 * q];
    }

    v8f acc0 = {0.f,0.f,0.f,0.f,0.f,0.f,0.f,0.f};
    v8f acc1 = {0.f,0.f,0.f,0.f,0.f,0.f,0.f,0.f};

    #pragma unroll
    for (int t = 0; t < TAPS; ++t) {
        const int cur = t & 1, nxt = cur ^ 1;
        if (t + 1 < TAPS) {          // prefetch next tap while current tap computes
            #pragma unroll
            for (int q = 0; q < 8; ++q) {
                A[nxt][q]  = *(const v4f*)&lds[aw_base + (t + 1) * 64 * STR + 4 * q];
                B0[nxt][q] = *(const v4f*)&lds[x0 + (t + 1) * STR + 4 * q];
                if constexpr (DUAL) B1[nxt][q] = *(const v4f*)&lds[x1 + (t + 1) * STR + 4 * q];
            }
        }
        #pragma unroll
        for (int q = 0; q < 8; ++q) {
            v2f a0 = {A[cur][q][0], A[cur][q][1]};     // K-chunk 2q   (this lane's pair)
            v2f a1 = {A[cur][q][2], A[cur][q][3]};     // K-chunk 2q+1
            v2f b0 = {B0[cur][q][0], B0[cur][q][1]};
            v2f b1 = {B0[cur][q][2], B0[cur][q][3]};
            acc0 = __builtin_amdgcn_wmma_f32_16x16x4_f32(false, a0, false, b0, (short)0, acc0, false, false);
            acc0 = __builtin_amdgcn_wmma_f32_16x16x4_f32(false, a1, false, b1, (short)0, acc0, false, false);
            if constexpr (DUAL) {
                v2f c0 = {B1[cur][q][0], B1[cur][q][1]};
                v2f c1 = {B1[cur][q][2], B1[cur][q][3]};
                acc1 = __builtin_amdgcn_wmma_f32_16x16x4_f32(false, a0, false, c0, (short)0, acc1, false, false);
                acc1 = __builtin_amdgcn_wmma_f32_16x16x4_f32(false, a1, false, c1, (short)0, acc1, false, false);
            }
        }
    }

    // D layout: VGPR r -> M = r + 8*lh, N = nm. Store res[pos][co] (two b128 each).
    int st0 = res_base + pt0 * 16 * STR;
    *(v4f*)&lds[st0]     = (v4f){acc0[0], acc0[1], acc0[2], acc0[3]};
    *(v4f*)&lds[st0 + 4] = (v4f){acc0[4], acc0[5], acc0[6], acc0[7]};
    if constexpr (DUAL) {
        int st1 = st0 + 16 * STR;
        *(v4f*)&lds[st1]     = (v4f){acc1[0], acc1[1], acc1[2], acc1[3]};
        *(v4f*)&lds[st1 + 4] = (v4f){acc1[4], acc1[5], acc1[6], acc1[7]};
    }
}

// Stage this residue's transposed weights, then compute all 9 position tiles.
template<int TAPS, int KOFS, int RHO>
__device__ __forceinline__ void process_residue(float* lds, const float* wgrp, int tid,
                                                int wrow, int aw_base, int xb_base,
                                                int res_nm_base)
{
    __syncthreads();   // x ready (first call) / previous residue's w reads done
    for (int e = tid; e < TAPS * 64 * 64; e += 256) {
        int t  = e >> 12;
        int r  = e & 4095;
        int co = r >> 6;
        int ci = r & 63;
        lds[W_OFF + (t * 64 + co) * STR + kslot(ci)] = wgrp[(co * 64 + ci) * 13 + 3 * t + KOFS];
    }
    __syncthreads();

    const int res_base = R_OFF0 + RHO * RES_SZ + res_nm_base;
    if (wrow == 0) {                                    // tiles 0..4
        gemm_tiles<TAPS, true >(lds, aw_base, xb_base, res_base, 0);
        gemm_tiles<TAPS, true >(lds, aw_base, xb_base, res_base, 2);
        gemm_tiles<TAPS, false>(lds, aw_base, xb_base, res_base, 4);
    } else {                                            // tiles 5..8
        gemm_tiles<TAPS, true >(lds, aw_base, xb_base, res_base, 5);
        gemm_tiles<TAPS, true >(lds, aw_base, xb_base, res_base, 7);
    }
}

__global__ __launch_bounds__(256) void trconv_fused_kernel(
    const float* __restrict__ x,      // [16][128][8192]
    const float* __restrict__ w,      // [128][64][13]
    const float* __restrict__ bias,   // [128]
    float* __restrict__ out)          // [16][128][8192]
{
    extern __shared__ float lds[];
    const int tid = threadIdx.x;
    const int b   = blockIdx.z;
    const int g   = blockIdx.y;                 // feature group (0..1), 64 ci / 64 co each
    const int p0  = blockIdx.x * NP;            // output position tile start
    const int me0 = p0 - 8;                     // residue frame start (halo for max/reflect)

    // ---- stage x tile: x_lds[lp][kslot(ci)] (garbage clamped rows never read)
    const float* xg = x + (size_t)(b * 128 + g * 64) * T_LEN;
    for (int e = tid; e < 64 * XROWS; e += 256) {
        int ci = e / XROWS;
        int lp = e - ci * XROWS;
        int m  = me0 + lp;
        m = m < 0 ? 0 : (m > T_LEN - 1 ? T_LEN - 1 : m);
        lds[XS_OFF + lp * STR + kslot(ci)] = xg[(size_t)ci * T_LEN + m];
    }
    if (tid < 64) lds[BIAS_OFF + tid] = bias[g * 64 + tid];

    // wave/lane geometry (wave32)
    const int lane = tid & 31;
    const int wv   = tid >> 5;       // 8 waves
    const int ct   = wv & 3;         // co tile (16 co each)
    const int wrow = wv >> 2;        // position-tile half
    const int nm   = lane & 15;      // M (A rows) / N (B cols)
    const int lh   = lane >> 4;      // lane half: K {0,1} vs {2,3} supplier

    const int aw_base     = W_OFF + (ct * 16 + nm) * STR + lh * 32;
    const int xb_base     = XS_OFF + nm * STR + lh * 32;
    const int res_nm_base = nm * STR + ct * 16 + lh * 8;

    const float* wgrp = w + (size_t)(g * 64) * (64 * 13);

    process_residue<4, 2, 0>(lds, wgrp, tid, wrow, aw_base, xb_base, res_nm_base); // res0: k=3t+2
    process_residue<4, 1, 1>(lds, wgrp, tid, wrow, aw_base, xb_base, res_nm_base); // res1: k=3t+1
    process_residue<5, 0, 2>(lds, wgrp, tid, wrow, aw_base, xb_base, res_nm_base); // res2: k=3t

    __syncthreads();

    // ---- fused epilogue ----
    const float* r0p = &lds[R_OFF0 + 0 * RES_SZ];
    const float* r1p = &lds[R_OFF0 + 1 * RES_SZ];
    const float* r2p = &lds[R_OFF0 + 2 * RES_SZ];
    for (int idx = tid; idx < 64 * NP; idx += 256) {
        int co = idx >> 7;               // NP == 128
        int il = idx & (NP - 1);
        int i  = p0 + il;
        float bi = lds[BIAS_OFF + co];

        int ri = ref_idx(i);
        int lm = ri - me0;
        float l0 = r0p[lm * STR + co], r0v = r0p[(lm + 1) * STR + co];
        float l1 = r1p[lm * STR + co], r1v = r1p[(lm + 1) * STR + co];
        float x_m = r2p[lm * STR + co] + bi;

        // 5-tap sliding max of biased res2 with ZERO padding outside [0, T)
        float um = x_m;                  // d == 0 term
        #pragma unroll
        for (int d = -2; d <= 2; ++d) {
            if (d == 0) continue;
            int j = i + d;
            float v = 0.0f;
            if (j >= 0 && j <= T_LEN - 1)
                v = r2p[(ref_idx(j) - me0) * STR + co] + bi;
            um = fmaxf(um, v);
        }

        float d0 = r0v - l0, d1 = r1v - l1;                   // bias cancels in gelu arg
        float qg0 = d0 / (1.0f + __expf(-1.702f * d0));
        float qg1 = d1 / (1.0f + __expf(-1.702f * d1));
        float accv = qg0 + qg1 + (r0v + l0 + r1v + l1) + 2.0f * x_m + 4.0f * bi + um;

        out[(size_t)(b * 128 + g * 64 + co) * T_LEN + i] = accv;
    }
}

extern "C" void kernel_launch(void* const* d_in, const int* in_sizes, int n_in,
                              void* d_out, int out_size, void* d_ws, size_t ws_size,
                              hipStream_t stream) {
    const float* x  = (const float*)d_in[0];
    const float* w  = (const float*)d_in[1];
    const float* b  = (const float*)d_in[2];
    float* out      = (float*)d_out;

    dim3 grid(T_LEN / NP, 2, NB);   // 64 pos-tiles x 2 groups x 16 batches = 2048 WGs
    dim3 block(256);                // 8 wave32
    size_t shmem = (size_t)LDS_FLOATS * sizeof(float);   // ~245 KB of the 320 KB WGP LDS
    hipLaunchKernelGGL(trconv_fused_kernel, grid, block, shmem, stream, x, w, b, out);
}